// MessagePassing_5471788335118
// MI455X (gfx1250) — compile-verified
//
#include <hip/hip_runtime.h>
#include <math.h>

typedef __attribute__((ext_vector_type(2))) float v2f;
typedef __attribute__((ext_vector_type(8))) float v8f;

#define H     128
#define NSRC  20000
#define NDST  20000
#define NEDGE 640000
#define MT    1250          // 20000 / 16 M-tiles

__device__ __forceinline__ float lrelu(float x) { return x > 0.0f ? x : 0.1f * x; }

// Single-hardware-atomic float max: signed atomicMax for non-negative values,
// unsigned atomicMin for negative values (IEEE-754 bit-order trick).
__device__ __forceinline__ void atomicMaxFloat(float* addr, float val) {
  if (val >= 0.0f) {
    atomicMax((int*)addr, __float_as_int(val));
  } else {
    atomicMin((unsigned int*)addr, (unsigned int)__float_as_int(val));
  }
}

// ---------------------------------------------------------------------------
// MLP GEMM: out[M,H] = f(A[M,H] @ W[H,H] + bias) (+ resid)
//   MODE 0: hidden = lrelu(src@W1 + b1)
//   MODE 1: h      = resid + hidden@W2 + b2
// One block = one 16-row M tile; wave w computes N columns [16w, 16w+16).
// A tile staged in LDS with stride 132 (conflict-free: 132 % 64 == 4).
// ---------------------------------------------------------------------------
template<int MODE>
__global__ void __launch_bounds__(256)
mlp_gemm_kernel(const float* __restrict__ A,
                const float* __restrict__ W,
                const float* __restrict__ bias,
                const float* __restrict__ resid,
                float* __restrict__ out) {
  __shared__ float At[16 * 132];
  const int mt   = blockIdx.x;
  const int tid  = threadIdx.x;
  const int wave = tid >> 5;
  const int lane = tid & 31;

  // Cooperative load of the 16x128 A tile (512 float4 slots, 2 per thread).
  {
    const float4* gsrc = (const float4*)(A + (size_t)mt * 16 * H);
    #pragma unroll
    for (int s = tid; s < 512; s += 256) {
      const int row = s >> 5;       // 32 float4 per row
      const int c4  = s & 31;
      float4 v = gsrc[row * 32 + c4];
      *(float4*)(&At[row * 132 + c4 * 4]) = v;
    }
  }
  __syncthreads();

  const int m  = lane & 15;         // A row for this lane
  const int kh = lane >> 4;         // K half: lanes 16-31 hold K+2
  const int nb = wave * 16;         // N tile base
  const int n  = lane & 15;         // B/C column within tile

  v8f c = {};
  const float* Arow = &At[m * 132 + 2 * kh];
  const float* Wcol = W + nb + n;

  #pragma unroll 8
  for (int kb = 0; kb < H; kb += 4) {
    v2f a; a.x = Arow[kb];                  a.y = Arow[kb + 1];
    v2f b; b.x = Wcol[(kb + 2 * kh) * H];   b.y = Wcol[(kb + 2 * kh + 1) * H];
    c = __builtin_amdgcn_wmma_f32_16x16x4_f32(false, a, false, b, (short)0, c,
                                              false, false);
  }

  const int   mBase = kh * 8;       // C/D: lanes 16-31 hold rows M+8
  const int   col   = nb + n;
  const float bv    = bias[col];
  #pragma unroll
  for (int r = 0; r < 8; ++r) {
    const int row = mt * 16 + mBase + r;
    float v = c[r] + bv;
    if (MODE == 0) v = lrelu(v);
    else           v += resid[(size_t)row * H + col];
    out[(size_t)row * H + col] = v;
  }
}

// ---------------------------------------------------------------------------
// Init accumulators: h_sum = 0, h_max = -inf, cnt = 0
// ---------------------------------------------------------------------------
__global__ void __launch_bounds__(256)
init_kernel(float* __restrict__ h_sum, float* __restrict__ h_max,
            float* __restrict__ cnt) {
  const int i = blockIdx.x * 256 + threadIdx.x;
  if (i < NDST * H) {
    h_sum[i] = 0.0f;
    h_max[i] = -__builtin_inff();
  }
  if (i < NDST) cnt[i] = 0.0f;
}

// ---------------------------------------------------------------------------
// Edge scatter: one wave per edge; each lane carries 4 features (float4).
// Gather h[src] (L2-resident, 10 MB) and atomically reduce into dst arrays.
// ---------------------------------------------------------------------------
__global__ void __launch_bounds__(256)
scatter_kernel(const float* __restrict__ h,
               const int* __restrict__ edge_src,
               const int* __restrict__ edge_dst,
               float* __restrict__ h_sum, float* __restrict__ h_max,
               float* __restrict__ cnt) {
  const int lane = threadIdx.x & 31;
  const int e    = (blockIdx.x * 256 + threadIdx.x) >> 5;
  if (e >= NEDGE) return;

  const int s = edge_src[e];
  const int d = edge_dst[e];
  const float4 v = *(const float4*)(h + (size_t)s * H + lane * 4);

  float* ps = h_sum + (size_t)d * H + lane * 4;
  float* pm = h_max + (size_t)d * H + lane * 4;
  atomicAdd(ps + 0, v.x); atomicAdd(ps + 1, v.y);
  atomicAdd(ps + 2, v.z); atomicAdd(ps + 3, v.w);
  atomicMaxFloat(pm + 0, v.x); atomicMaxFloat(pm + 1, v.y);
  atomicMaxFloat(pm + 2, v.z); atomicMaxFloat(pm + 3, v.w);
  if (lane == 0) atomicAdd(cnt + d, 1.0f);
}

// ---------------------------------------------------------------------------
// Readout GEMM: out = dst_feat + lrelu([sum|max*|mean|dst] @ Wr + br)
// The 16x512 concat tile is built directly in LDS (stride 516, conflict-free),
// then 128 K-steps of f32 WMMA against Wr[512,128].
// ---------------------------------------------------------------------------
__global__ void __launch_bounds__(256)
final_gemm_kernel(const float* __restrict__ h_sum,
                  const float* __restrict__ h_max,
                  const float* __restrict__ cnt,
                  const float* __restrict__ dst_feat,
                  const float* __restrict__ Wr,
                  const float* __restrict__ br,
                  float* __restrict__ out) {
  __shared__ float At[16 * 516];
  const int mt   = blockIdx.x;
  const int tid  = threadIdx.x;
  const int wave = tid >> 5;
  const int lane = tid & 31;

  // Build A tile (16 rows x 512 cols): 2048 float4 slots, 8 per thread.
  #pragma unroll
  for (int s = tid; s < 2048; s += 256) {
    const int row  = s >> 7;          // 128 float4 per row
    const int c4   = s & 127;
    const int col  = c4 * 4;
    const int seg  = col >> 7;        // 0:sum 1:max 2:mean 3:dst
    const int kk   = col & 127;
    const int grow = mt * 16 + row;
    const float cn = cnt[grow];
    float4 v;
    if (seg == 0) {
      v = *(const float4*)(h_sum + (size_t)grow * H + kk);
    } else if (seg == 1) {
      v = *(const float4*)(h_max + (size_t)grow * H + kk);
      if (!(cn > 0.0f)) v = make_float4(0.0f, 0.0f, 0.0f, 0.0f);
    } else if (seg == 2) {
      v = *(const float4*)(h_sum + (size_t)grow * H + kk);
      const float inv = 1.0f / fmaxf(cn, 1.0f);
      v.x *= inv; v.y *= inv; v.z *= inv; v.w *= inv;
    } else {
      v = *(const float4*)(dst_feat + (size_t)grow * H + kk);
    }
    *(float4*)(&At[row * 516 + col]) = v;
  }
  __syncthreads();

  const int m  = lane & 15;
  const int kh = lane >> 4;
  const int nb = wave * 16;
  const int n  = lane & 15;

  v8f c = {};
  const float* Arow = &At[m * 516 + 2 * kh];
  const float* Wcol = Wr + nb + n;

  #pragma unroll 8
  for (int kb = 0; kb < 512; kb += 4) {
    v2f a; a.x = Arow[kb];                  a.y = Arow[kb + 1];
    v2f b; b.x = Wcol[(kb + 2 * kh) * H];   b.y = Wcol[(kb + 2 * kh + 1) * H];
    c = __builtin_amdgcn_wmma_f32_16x16x4_f32(false, a, false, b, (short)0, c,
                                              false, false);
  }

  const int   mBase = kh * 8;
  const int   col   = nb + n;
  const float bv    = br[col];
  #pragma unroll
  for (int r = 0; r < 8; ++r) {
    const int row = mt * 16 + mBase + r;
    const float res = dst_feat[(size_t)row * H + col];
    out[(size_t)row * H + col] = res + lrelu(c[r] + bv);
  }
}

// ---------------------------------------------------------------------------
extern "C" void kernel_launch(void* const* d_in, const int* in_sizes, int n_in,
                              void* d_out, int out_size, void* d_ws, size_t ws_size,
                              hipStream_t stream) {
  const float* src_feat = (const float*)d_in[0];
  const float* dst_feat = (const float*)d_in[1];
  const int*   edge_src = (const int*)d_in[2];
  const int*   edge_dst = (const int*)d_in[3];
  const float* W1       = (const float*)d_in[4];
  const float* b1       = (const float*)d_in[5];
  const float* W2       = (const float*)d_in[6];
  const float* b2       = (const float*)d_in[7];
  const float* Wr       = (const float*)d_in[8];
  const float* br       = (const float*)d_in[9];
  float*       out      = (float*)d_out;

  // Workspace layout (floats): hidden | h | h_sum | h_max | cnt  (~41 MB)
  float* ws     = (float*)d_ws;
  float* hidden = ws;
  float* h      = hidden + (size_t)NSRC * H;
  float* h_sum  = h      + (size_t)NSRC * H;
  float* h_max  = h_sum  + (size_t)NDST * H;
  float* cntp   = h_max  + (size_t)NDST * H;

  const dim3 blk(256);

  // 1) hidden = lrelu(src @ W1 + b1)
  mlp_gemm_kernel<0><<<MT, blk, 0, stream>>>(src_feat, W1, b1, nullptr, hidden);
  // 2) h = src + hidden @ W2 + b2
  mlp_gemm_kernel<1><<<MT, blk, 0, stream>>>(hidden, W2, b2, src_feat, h);
  // 3) zero/neg-inf accumulators
  init_kernel<<<(NDST * H + 255) / 256, blk, 0, stream>>>(h_sum, h_max, cntp);
  // 4) edge gather + fused sum/max/count scatter (one wave per edge)
  scatter_kernel<<<(NEDGE * 32 + 255) / 256, blk, 0, stream>>>(
      h, edge_src, edge_dst, h_sum, h_max, cntp);
  // 5) out = dst + lrelu(concat @ Wr + br)
  final_gemm_kernel<<<MT, blk, 0, stream>>>(h_sum, h_max, cntp, dst_feat, Wr, br, out);
}